// JetGAT_67808943669847
// MI455X (gfx1250) — compile-verified
//
#include <hip/hip_runtime.h>
#include <hip/hip_bf16.h>

#define NNODES 50000
#define NEDGES 1600000
#define NGRAPH 2000

typedef __attribute__((ext_vector_type(2))) float v2f;
typedef __attribute__((ext_vector_type(8))) float v8f;

// -inf encoded with the order-preserving float->uint key (for atomic max)
#define NEGINF_KEY 0x007FFFFFu

__device__ __forceinline__ float lrelu(float x) { return x > 0.f ? x : 0.2f * x; }
__device__ __forceinline__ float eluf(float x)  { return x > 0.f ? x : expm1f(x); }

__device__ __forceinline__ unsigned f2key(float f) {
  unsigned u = __float_as_uint(f);
  return (u & 0x80000000u) ? ~u : (u | 0x80000000u);
}
__device__ __forceinline__ float key2f(unsigned k) {
  return __uint_as_float((k & 0x80000000u) ? (k & 0x7fffffffu) : ~k);
}

// ---------------------------------------------------------------- fill kernels
__global__ void fill_f32(float* __restrict__ p, float v, int n) {
  int i = blockIdx.x * blockDim.x + threadIdx.x;
  if (i < n) p[i] = v;
}
__global__ void fill_u32(unsigned* __restrict__ p, unsigned v, int n) {
  int i = blockIdx.x * blockDim.x + threadIdx.x;
  if (i < n) p[i] = v;
}

// ------------------------------------------------- layer-1 GEMM (K=3, scalar)
__global__ void gemm_small(const float* __restrict__ A, const float* __restrict__ B,
                           float* __restrict__ C, int n, int K, int Nc) {
  int i = blockIdx.x * blockDim.x + threadIdx.x;
  if (i >= n * Nc) return;
  int r = i / Nc, c = i % Nc;
  float s = 0.f;
  for (int k = 0; k < K; ++k) s += A[r * K + k] * B[k * Nc + c];
  C[i] = s;
}

// ------------------------------------- fp32 WMMA GEMM: C[M,Nc] = A[M,K]*B[K,Nc]
// One wave per 16x(16*TN) strip.  M%16==0, K%4==0, Nc%(16*TN)==0 guaranteed.
// A 16x4 layout: lanes 0-15 carry K={k0,k0+1}, lanes 16-31 carry K={k0+2,k0+3}.
// C/D 16x16 layout: VGPR r -> row r (lanes 0-15) / row r+8 (lanes 16-31).
template <int TN>
__global__ void wmma_gemm_f32(const float* __restrict__ A, const float* __restrict__ B,
                              float* __restrict__ C, int K, int Nc) {
  int lane = threadIdx.x & 31;
  int half = lane >> 4;       // 0 or 1
  int l16  = lane & 15;
  int rowA    = blockIdx.x * 16 + l16;
  int colBase = blockIdx.y * (16 * TN);

  v8f acc[TN] = {};
  for (int k0 = 0; k0 < K; k0 += 4) {
    int ka = k0 + 2 * half;
    v2f a;
    a.x = A[(size_t)rowA * K + ka];
    a.y = A[(size_t)rowA * K + ka + 1];
#pragma unroll
    for (int t = 0; t < TN; ++t) {
      int col = colBase + t * 16 + l16;
      v2f b;
      b.x = B[(size_t)ka * Nc + col];
      b.y = B[(size_t)(ka + 1) * Nc + col];
      acc[t] = __builtin_amdgcn_wmma_f32_16x16x4_f32(
          /*neg_a=*/false, a, /*neg_b=*/false, b,
          /*c_mod=*/(short)0, acc[t], /*reuse_a=*/false, /*reuse_b=*/false);
    }
  }
  int row0 = blockIdx.x * 16 + half * 8;
#pragma unroll
  for (int t = 0; t < TN; ++t) {
    int col = colBase + t * 16 + l16;
#pragma unroll
    for (int r = 0; r < 8; ++r)
      C[(size_t)(row0 + r) * Nc + col] = acc[t][r];
  }
}

// ------------------------------------------- per-node attention scores es/ed
__global__ void score_kernel(const float* __restrict__ h, const float* __restrict__ a_s,
                             const float* __restrict__ a_d, float* __restrict__ es,
                             float* __restrict__ ed, int n, int H, int C) {
  int i = blockIdx.x * blockDim.x + threadIdx.x;
  if (i >= n * H) return;
  int node = i / H, head = i % H;
  const float* hp = h + (size_t)node * H * C + (size_t)head * C;
  float s = 0.f, d = 0.f;
  for (int c = 0; c < C; ++c) {
    float v = hp[c];
    s += v * a_s[head * C + c];
    d += v * a_d[head * C + c];
  }
  es[i] = s;
  ed[i] = d;
}

// ------------------------------------------- edge pass 1: segment max (exact)
__global__ void edge_max(const int* __restrict__ ei, const float* __restrict__ es,
                         const float* __restrict__ ed, unsigned* __restrict__ mkey, int H) {
  int i = blockIdx.x * blockDim.x + threadIdx.x;
  int total = (NEDGES + NNODES) * H;
  if (i >= total) return;
  int eid = i / H, head = i % H;
  int s = (eid < NEDGES) ? ei[eid] : (eid - NEDGES);
  int d = (eid < NEDGES) ? ei[NEDGES + eid] : (eid - NEDGES);
  float e = lrelu(es[s * H + head] + ed[d * H + head]);
  atomicMax(&mkey[d * H + head], f2key(e));
}

// ------------------------------------------- edge pass 2: segment sum of exp
__global__ void edge_sum(const int* __restrict__ ei, const float* __restrict__ es,
                         const float* __restrict__ ed, const unsigned* __restrict__ mkey,
                         float* __restrict__ z, int H) {
  int i = blockIdx.x * blockDim.x + threadIdx.x;
  int total = (NEDGES + NNODES) * H;
  if (i >= total) return;
  int eid = i / H, head = i % H;
  int s = (eid < NEDGES) ? ei[eid] : (eid - NEDGES);
  int d = (eid < NEDGES) ? ei[NEDGES + eid] : (eid - NEDGES);
  float e = lrelu(es[s * H + head] + ed[d * H + head]);
  float m = key2f(mkey[d * H + head]);
  atomicAdd(&z[d * H + head], expf(e - m));
}

// --------------------------------- edge pass 3: alpha-weighted scatter of h[src]
__global__ void edge_agg(const int* __restrict__ ei, const float* __restrict__ es,
                         const float* __restrict__ ed, const unsigned* __restrict__ mkey,
                         const float* __restrict__ z, const float* __restrict__ h,
                         float* __restrict__ agg, int H, int C) {
  int i = blockIdx.x * blockDim.x + threadIdx.x;
  int HC = H * C;
  int total = (NEDGES + NNODES) * HC;
  if (i >= total) return;
  int eid = i / HC, ch = i % HC;
  int head = ch / C;
  int s = (eid < NEDGES) ? ei[eid] : (eid - NEDGES);
  int d = (eid < NEDGES) ? ei[NEDGES + eid] : (eid - NEDGES);
  float e = lrelu(es[s * H + head] + ed[d * H + head]);
  float m = key2f(mkey[d * H + head]);
  float alpha = expf(e - m) / (z[d * H + head] + 1e-16f);
  atomicAdd(&agg[(size_t)d * HC + ch], h[(size_t)s * HC + ch] * alpha);
}

// ------------------------------------------- bias + ELU, in place
__global__ void finalize_kernel(float* __restrict__ F, const float* __restrict__ b,
                                int total, int HC) {
  int i = blockIdx.x * blockDim.x + threadIdx.x;
  if (i >= total) return;
  F[i] = eluf(F[i] + b[i % HC]);
}

// ------------------------------------------- graph mean-pool (accumulate)
__global__ void pool_kernel(const float* __restrict__ feat, const int* __restrict__ batch,
                            float* __restrict__ pooled, float* __restrict__ cnt, int C) {
  int i = blockIdx.x * blockDim.x + threadIdx.x;
  if (i >= NNODES * C) return;
  int node = i / C, c = i % C;
  int g = batch[node];
  atomicAdd(&pooled[g * C + c], feat[(size_t)node * C + c]);
  if (c == 0) atomicAdd(&cnt[g], 1.0f);
}

// ------------------------------------------- pooled @ Wh + bh
__global__ void head_kernel(const float* __restrict__ pooled, const float* __restrict__ cnt,
                            const float* __restrict__ Wh, const float* __restrict__ bh,
                            float* __restrict__ out) {
  int i = blockIdx.x * blockDim.x + threadIdx.x;
  if (i >= NGRAPH * 2) return;
  int g = i / 2, k = i % 2;
  float c = cnt[g];
  c = c > 1.0f ? c : 1.0f;
  float s = 0.f;
  for (int j = 0; j < 32; ++j) s += (pooled[g * 32 + j] / c) * Wh[j * 2 + k];
  out[i] = s + bh[k];
}

// =============================================================================
extern "C" void kernel_launch(void* const* d_in, const int* in_sizes, int n_in,
                              void* d_out, int out_size, void* d_ws, size_t ws_size,
                              hipStream_t stream) {
  (void)in_sizes; (void)n_in; (void)out_size; (void)ws_size;

  const float* x     = (const float*)d_in[0];
  const int*   ei    = (const int*)d_in[1];
  const int*   batch = (const int*)d_in[2];
  const float* W1  = (const float*)d_in[3];
  const float* a1s = (const float*)d_in[4];
  const float* a1d = (const float*)d_in[5];
  const float* b1  = (const float*)d_in[6];
  const float* W2  = (const float*)d_in[7];
  const float* a2s = (const float*)d_in[8];
  const float* a2d = (const float*)d_in[9];
  const float* b2  = (const float*)d_in[10];
  const float* W3  = (const float*)d_in[11];
  const float* a3s = (const float*)d_in[12];
  const float* a3d = (const float*)d_in[13];
  const float* b3  = (const float*)d_in[14];
  const float* Wh  = (const float*)d_in[15];
  const float* bh  = (const float*)d_in[16];
  float* out = (float*)d_out;

  // workspace carve-up (floats)
  float*    F      = (float*)d_ws;                      // [N,128] activations / agg
  float*    Hb     = F  + (size_t)NNODES * 128;         // [N,128] transformed feats
  float*    es     = Hb + (size_t)NNODES * 128;         // [N,4]
  float*    ed     = es + (size_t)NNODES * 4;           // [N,4]
  unsigned* mk     = (unsigned*)(ed + (size_t)NNODES * 4);  // [N,4]
  float*    z      = (float*)(mk + (size_t)NNODES * 4);     // [N,4]
  float*    pooled = z + (size_t)NNODES * 4;            // [G,32]
  float*    cnt    = pooled + (size_t)NGRAPH * 32;      // [G]

  const int T = 256;
  auto blk = [](int n) { return (n + 255) / 256; };

  const int EN = NEDGES + NNODES;

  // ---------------- Layer 1 : K=3 -> [N,4,32], concat -----------------------
  {
    const int H = 4, C = 32, HC = 128;
    gemm_small<<<blk(NNODES * HC), T, 0, stream>>>(x, W1, Hb, NNODES, 3, HC);
    score_kernel<<<blk(NNODES * H), T, 0, stream>>>(Hb, a1s, a1d, es, ed, NNODES, H, C);
    fill_u32<<<blk(NNODES * H), T, 0, stream>>>(mk, NEGINF_KEY, NNODES * H);
    fill_f32<<<blk(NNODES * H), T, 0, stream>>>(z, 0.f, NNODES * H);
    fill_f32<<<blk(NNODES * HC), T, 0, stream>>>(F, 0.f, NNODES * HC);
    edge_max<<<blk(EN * H), T, 0, stream>>>(ei, es, ed, mk, H);
    edge_sum<<<blk(EN * H), T, 0, stream>>>(ei, es, ed, mk, z, H);
    edge_agg<<<blk(EN * HC), T, 0, stream>>>(ei, es, ed, mk, z, Hb, F, H, C);
    finalize_kernel<<<blk(NNODES * HC), T, 0, stream>>>(F, b1, NNODES * HC, HC);
  }

  // ---------------- Layer 2 : [N,128]x[128,128] WMMA, concat ----------------
  {
    const int H = 4, C = 32, HC = 128;
    dim3 grid(NNODES / 16, HC / 64);   // TN=4 -> 64 cols per wave
    wmma_gemm_f32<4><<<grid, dim3(32), 0, stream>>>(F, W2, Hb, 128, HC);
    score_kernel<<<blk(NNODES * H), T, 0, stream>>>(Hb, a2s, a2d, es, ed, NNODES, H, C);
    fill_u32<<<blk(NNODES * H), T, 0, stream>>>(mk, NEGINF_KEY, NNODES * H);
    fill_f32<<<blk(NNODES * H), T, 0, stream>>>(z, 0.f, NNODES * H);
    fill_f32<<<blk(NNODES * HC), T, 0, stream>>>(F, 0.f, NNODES * HC);
    edge_max<<<blk(EN * H), T, 0, stream>>>(ei, es, ed, mk, H);
    edge_sum<<<blk(EN * H), T, 0, stream>>>(ei, es, ed, mk, z, H);
    edge_agg<<<blk(EN * HC), T, 0, stream>>>(ei, es, ed, mk, z, Hb, F, H, C);
    finalize_kernel<<<blk(NNODES * HC), T, 0, stream>>>(F, b2, NNODES * HC, HC);
  }

  // ---------------- Layer 3 : [N,128]x[128,32] WMMA, H=1 (mean==identity) ---
  {
    const int H = 1, C = 32, HC = 32;
    dim3 grid(NNODES / 16, 1);         // TN=2 -> 32 cols per wave
    wmma_gemm_f32<2><<<grid, dim3(32), 0, stream>>>(F, W3, Hb, 128, HC);
    score_kernel<<<blk(NNODES * H), T, 0, stream>>>(Hb, a3s, a3d, es, ed, NNODES, H, C);
    fill_u32<<<blk(NNODES * H), T, 0, stream>>>(mk, NEGINF_KEY, NNODES * H);
    fill_f32<<<blk(NNODES * H), T, 0, stream>>>(z, 0.f, NNODES * H);
    fill_f32<<<blk(NNODES * HC), T, 0, stream>>>(F, 0.f, NNODES * HC);
    edge_max<<<blk(EN * H), T, 0, stream>>>(ei, es, ed, mk, H);
    edge_sum<<<blk(EN * H), T, 0, stream>>>(ei, es, ed, mk, z, H);
    edge_agg<<<blk(EN * HC), T, 0, stream>>>(ei, es, ed, mk, z, Hb, F, H, C);
    finalize_kernel<<<blk(NNODES * HC), T, 0, stream>>>(F, b3, NNODES * HC, HC);
  }

  // ---------------- mean pool + linear head ---------------------------------
  fill_f32<<<blk(NGRAPH * 32), T, 0, stream>>>(pooled, 0.f, NGRAPH * 32);
  fill_f32<<<blk(NGRAPH), T, 0, stream>>>(cnt, 0.f, NGRAPH);
  pool_kernel<<<blk(NNODES * 32), T, 0, stream>>>(F, batch, pooled, cnt, 32);
  head_kernel<<<blk(NGRAPH * 2), T, 0, stream>>>(pooled, cnt, Wh, bh, out);
}